// Aggregate_87866440942142
// MI455X (gfx1250) — compile-verified
//
#include <hip/hip_runtime.h>
#include <hip/hip_bf16.h>

#define N_NODES 100000
#define D 128

typedef __attribute__((ext_vector_type(2))) float v2f;
typedef __attribute__((ext_vector_type(8))) float v8f;

// ---------------------------------------------------------------------------
// Kernel 1: zero the accumulator buffer y [N_NODES * D] floats in d_ws
// ---------------------------------------------------------------------------
__global__ void zero_y_kernel(float4* __restrict__ y4, int n4) {
    int i = blockIdx.x * blockDim.x + threadIdx.x;
    if (i < n4) y4[i] = make_float4(0.f, 0.f, 0.f, 0.f);
}

// ---------------------------------------------------------------------------
// Kernel 2: COO SpMM scatter-add.
// One wave32 per edge. Lane L handles features [4L, 4L+4).
// x fits in L2 (51 MB < 192 MB) so the gather is L2-resident;
// atomicAdd(float) lowers to GLOBAL_ATOMIC_ADD_F32 (no-return, STOREcnt).
// ---------------------------------------------------------------------------
__global__ void spmm_kernel(const float* __restrict__ x,
                            const long long* __restrict__ erow,
                            const long long* __restrict__ ecol,
                            const float* __restrict__ eval,
                            float* __restrict__ y,
                            int n_edges) {
    const int wave = blockIdx.x * (blockDim.x >> 5) + (threadIdx.x >> 5);
    if (wave >= n_edges) return;
    const int lane = threadIdx.x & 31;

    const long long r = erow[wave];
    const long long c = ecol[wave];
    const float    v = eval[wave];

    const float4 m = *((const float4*)(x + c * D) + lane);  // coalesced 512B/wave
    float* yd = y + r * D + lane * 4;
    atomicAdd(yd + 0, m.x * v);
    atomicAdd(yd + 1, m.y * v);
    atomicAdd(yd + 2, m.z * v);
    atomicAdd(yd + 3, m.w * v);
}

// ---------------------------------------------------------------------------
// Kernel 3: out = y @ W^T via V_WMMA_F32_16X16X4_F32 (full fp32 precision).
// One wave per 16x16 output tile; 100000 = 6250*16 rows, 128 = 8*16 cols
// -> 50000 tiles, 8 waves/block, 6250 blocks (exact cover: EXEC all-1s).
//
// ISA f32 operand layouts (CDNA5 ISA §7.12.2):
//   A 16x4 : lane = M (0..15 twice); VGPR0/1 = K pair; lane-half picks K 0,1 vs 2,3
//   B 4x16 : lane = N; same K striping (B[k][n] = W[col0+n][k], i.e. W^T tile)
//   C/D    : VGPR r = row M=r (lanes 0-15) / M=r+8 (lanes 16-31), lane = N
// ---------------------------------------------------------------------------
__global__ void gemm_wmma_kernel(const float* __restrict__ y,
                                 const float* __restrict__ W,
                                 float* __restrict__ out) {
    const int lane  = threadIdx.x & 31;
    const int wavid = threadIdx.x >> 5;
    const int tile  = blockIdx.x * 8 + wavid;   // 0..49999
    const int row0  = (tile >> 3) * 16;         // row tile * 16
    const int col0  = (tile & 7) * 16;          // col tile * 16

    const int mn    = lane & 15;
    const int kpair = (lane >> 4) * 2;          // 0 or 2

    const float* aptr = y + (long long)(row0 + mn) * D + kpair;  // A: y rows
    const float* bptr = W + (long long)(col0 + mn) * D + kpair;  // B: W^T cols

    v8f acc = {};
#pragma unroll
    for (int k0 = 0; k0 < D; k0 += 4) {
        v2f a, b;
        a.x = aptr[k0];
        a.y = aptr[k0 + 1];
        b.x = bptr[k0];
        b.y = bptr[k0 + 1];
        // 8 args: (neg_a, A, neg_b, B, c_mod, C, reuse_a, reuse_b)
        acc = __builtin_amdgcn_wmma_f32_16x16x4_f32(
            false, a, false, b, (short)0, acc, false, false);
    }

    const int n     = lane & 15;
    const int mbase = (lane >> 4) * 8;
#pragma unroll
    for (int r = 0; r < 8; ++r) {
        out[(long long)(row0 + mbase + r) * D + col0 + n] = acc[r];
    }
}

// ---------------------------------------------------------------------------
// Launch
// ---------------------------------------------------------------------------
extern "C" void kernel_launch(void* const* d_in, const int* in_sizes, int n_in,
                              void* d_out, int out_size, void* d_ws, size_t ws_size,
                              hipStream_t stream) {
    const float*     x    = (const float*)d_in[0];      // [N_NODES, D] f32
    const long long* erow = (const long long*)d_in[1];  // [E] i64
    const long long* ecol = (const long long*)d_in[2];  // [E] i64
    const float*     eval = (const float*)d_in[3];      // [E] f32
    const float*     W    = (const float*)d_in[4];      // [D, D] f32
    float*           out  = (float*)d_out;              // [N_NODES, D] f32
    float*           y    = (float*)d_ws;               // [N_NODES, D] f32 scratch

    const int n_edges = in_sizes[3];

    // 1) y = 0
    {
        const int n4 = (N_NODES * D) / 4;
        dim3 blk(256), grd((n4 + 255) / 256);
        zero_y_kernel<<<grd, blk, 0, stream>>>((float4*)y, n4);
    }

    // 2) scatter-add SpMM: one wave per edge, 8 waves (256 threads) per block
    {
        const int waves_per_block = 8;
        dim3 blk(256), grd((n_edges + waves_per_block - 1) / waves_per_block);
        spmm_kernel<<<grd, blk, 0, stream>>>(x, erow, ecol, eval, y, n_edges);
    }

    // 3) out = y @ W^T with fp32 WMMA; 50000 tiles, 8 waves/block -> 6250 blocks
    {
        dim3 blk(256), grd((N_NODES / 16) * (D / 16) / 8);
        gemm_wmma_kernel<<<grd, blk, 0, stream>>>(y, W, out);
    }
}